// GramSchmidt_21844203667976
// MI455X (gfx1250) — compile-verified
//
#include <hip/hip_runtime.h>

#define EPS 1e-8f

typedef __attribute__((ext_vector_type(2))) float v2f;
typedef __attribute__((ext_vector_type(8))) float v8f;
typedef __attribute__((ext_vector_type(4))) unsigned int u32x4;
typedef __attribute__((ext_vector_type(8))) int i32x8;
typedef __attribute__((ext_vector_type(4))) int i32x4;

#define NROW     256
#define DIM      1024
#define PANEL    16
#define NPANELS  (NROW / PANEL)      // 16
#define NTHREADS 512
#define NWAVES   (NTHREADS / 32)     // 16 wave32s

// LDS byte offset of a __shared__ object: flat LDS addresses carry the
// aperture in the high 32 bits, offset in the low 32 bits.
__device__ __forceinline__ unsigned lds_off(const void* p) {
  return (unsigned)(unsigned long long)(size_t)p;
}

// Issue a TDM load of a PANEL x DIM fp32 tile (row-major, stride DIM) from
// global memory into LDS. 2-D descriptor per CDNA5 ISA.
__device__ __forceinline__ void tdm_load_panel(const float* gsrc, unsigned ldsaddr) {
  const unsigned long long ga = (unsigned long long)(size_t)gsrc;
  u32x4 g0;
  g0[0] = 1u;                                        // count=1, user descriptor
  g0[1] = ldsaddr;                                   // lds_addr [63:32]
  g0[2] = (unsigned)(ga & 0xFFFFFFFFull);            // global_addr lo
  g0[3] = (unsigned)((ga >> 32) & 0x01FFFFFFull)     // global_addr [56:32]
        | (2u << 30);                                // type = 2 ("image")
  i32x8 g1;
  g1[0] = (int)(2u << 16);                           // data_size = 4 bytes
  g1[1] = (int)(((unsigned)DIM & 0xFFFFu) << 16);    // tensor_dim0 lo16 @63:48
  g1[2] = (int)((((unsigned)DIM >> 16) & 0xFFFFu)    // tensor_dim0 hi16 @79:64
        | (((unsigned)PANEL & 0xFFFFu) << 16));      // tensor_dim1 lo16 @95:80
  g1[3] = (int)(((unsigned)DIM & 0xFFFFu) << 16);    // tile_dim0 @127:112 (dim1 hi16 = 0)
  g1[4] = (int)(PANEL & 0xFFFF);                     // tile_dim1 @143:128 (tile_dim2 = 0)
  g1[5] = (int)DIM;                                  // tensor_dim0_stride lo32 @191:160
  g1[6] = 0;                                         // stride0 hi16 | stride1 lo16
  g1[7] = 0;                                         // stride1 hi32
  i32x4 z4 = {};                                     // groups 2/3: 2-D tile, unused dims
  i32x8 z8 = {};                                     // trailing group (clang-23 6-arg form)
  __builtin_amdgcn_tensor_load_to_lds(g0, g1, z4, z4, z8, 0);
}

__launch_bounds__(NTHREADS)
__global__ void gram_schmidt_kernel(const float* __restrict__ Vg_all,
                                    float* __restrict__ Qg_all) {
  const int batch = blockIdx.x;
  const float* Vg = Vg_all + (size_t)batch * NROW * DIM;
  float*       Qg = Qg_all + (size_t)batch * NROW * DIM;

  // 2x64KB (V ping-pong, TDM double buffer) + 64KB (U) + 16KB (C) ~= 209KB
  __shared__ float sVbuf[2][PANEL * DIM];  // original panel rows (TDM dest)
  __shared__ float sU[PANEL * DIM];        // residual / q accumulator
  __shared__ float sC[PANEL * NROW];       // scaled projection coefficients
  __shared__ float sNs[NROW];              // q.q for each finished row
  __shared__ float sDot[PANEL];            // intra-panel projection coefs
  __shared__ float sRed[NWAVES];           // block reduction scratch

  const int tid  = threadIdx.x;
  const int wave = tid >> 5;
  const int lane = tid & 31;
  const int m    = lane & 15;   // M (or N) index inside a 16-wide WMMA tile
  const int hi   = lane >> 4;   // lane half select (K split per ISA layout)

  // Prologue: kick off the DMA for panel 0 while nothing else is in flight.
  if (wave == 0)
    tdm_load_panel(Vg, lds_off(&sVbuf[0][0]));

  for (int p = 0; p < NPANELS; ++p) {
    const int r0 = p * PANEL;   // first global row of this panel
    const int J  = r0;          // number of completed q rows
    float* sV = sVbuf[p & 1];   // current panel's original rows

    // ---- A1: land the in-flight TDM tile, then prefetch the next panel ----
    if (wave == 0)
      __builtin_amdgcn_s_wait_tensorcnt(0);
    __syncthreads();                         // sV now valid for all waves
    if (wave == 0 && p + 1 < NPANELS)        // overlap next DMA with compute
      tdm_load_panel(Vg + (size_t)(r0 + PANEL) * DIM,
                     lds_off(&sVbuf[(p + 1) & 1][0]));
    // working copy U = V (LDS -> LDS through registers)
    for (int t = tid; t < PANEL * DIM / 4; t += NTHREADS)
      ((float4*)sU)[t] = ((const float4*)sV)[t];
    __syncthreads();

    // ---- A2: dots GEMM  C(16 x J) = panelV (16 x D) * Qprev^T (D x J) ----
    // Each wave owns one 16x16 tile of C (16 previous q rows).
    for (int jt = wave; jt < p; jt += NWAVES) {
      const int j0 = jt * PANEL;
      __builtin_prefetch(&Qg[(size_t)(j0 + m) * DIM], 0, 0);
      v8f acc = {};
      for (int k = 0; k < DIM; k += 4) {
        const int klo = k + hi * 2;
        v2f a, b;
        // A: 16x4 fp32 tile of panelV (M=m, K split across lane halves)
        a.x = sV[m * DIM + klo];
        a.y = sV[m * DIM + klo + 1];
        // B: 4x16 tile of Qprev^T -> B[kk][n] = Q[j0+n][k+kk]
        b.x = Qg[(size_t)(j0 + m) * DIM + klo];
        b.y = Qg[(size_t)(j0 + m) * DIM + klo + 1];
        acc = __builtin_amdgcn_wmma_f32_16x16x4_f32(
            false, a, false, b, (short)0, acc, false, false);
      }
      // scale by 1/(q.q + eps) and stash coefficients
      #pragma unroll
      for (int r = 0; r < 8; ++r) {
        const int mi = hi * 8 + r;        // D-matrix row per ISA layout
        const int j  = j0 + m;            // D-matrix col = previous row id
        sC[mi * NROW + j] = acc[r] / (sNs[j] + EPS);
      }
    }
    __syncthreads();

    // ---- A3: update GEMM  panelU -= C (16 x J) * Qprev (J x D) ----
    // Each wave owns disjoint 16-column tiles of d.
    if (p > 0) {
      for (int ct = wave; ct < DIM / PANEL; ct += NWAVES) {
        const int c0 = ct * PANEL;
        v8f acc = {};
        for (int j = 0; j < J; j += 4) {
          const int jlo = j + hi * 2;
          v2f a, b;
          // A: 16x4 tile of coefficients (M=m, K=j..j+3)
          a.x = sC[m * NROW + jlo];
          a.y = sC[m * NROW + jlo + 1];
          // B: 4x16 tile of Qprev -> B[kk][n] = Q[j+kk][c0+n]; coalesced rows
          b.x = Qg[(size_t)jlo * DIM + c0 + m];
          b.y = Qg[(size_t)(jlo + 1) * DIM + c0 + m];
          acc = __builtin_amdgcn_wmma_f32_16x16x4_f32(
              false, a, false, b, (short)0, acc, false, false);
        }
        #pragma unroll
        for (int r = 0; r < 8; ++r) {
          const int mi = hi * 8 + r;
          sU[mi * DIM + c0 + m] -= acc[r];
        }
      }
    }
    __syncthreads();

    // ---- A4: intra-panel classical Gram-Schmidt (16 sequential rows) ----
    for (int i = 0; i < PANEL; ++i) {
      // dots of ORIGINAL v_i with panel-local finished rows j' < i
      if (wave < i) {
        float part = 0.f;
        for (int k = lane; k < DIM; k += 32)
          part += sU[wave * DIM + k] * sV[i * DIM + k];
        #pragma unroll
        for (int off = 16; off > 0; off >>= 1)
          part += __shfl_down(part, off, 32);
        if (lane == 0)
          sDot[wave] = part / (sNs[r0 + wave] + EPS);
      }
      __syncthreads();

      // subtract projections from u_i
      for (int k = tid; k < DIM; k += NTHREADS) {
        float u = sU[i * DIM + k];
        for (int j2 = 0; j2 < i; ++j2)
          u -= sDot[j2] * sU[j2 * DIM + k];
        sU[i * DIM + k] = u;
      }
      __syncthreads();

      // norm of u_i (block reduction)
      float part = 0.f;
      for (int k = tid; k < DIM; k += NTHREADS) {
        float u = sU[i * DIM + k];
        part += u * u;
      }
      #pragma unroll
      for (int off = 16; off > 0; off >>= 1)
        part += __shfl_down(part, off, 32);
      if (lane == 0) sRed[wave] = part;
      __syncthreads();
      if (tid == 0) {
        float s = 0.f;
        #pragma unroll
        for (int wv = 0; wv < NWAVES; ++wv) s += sRed[wv];
        const float nrm   = sqrtf(s);
        const float scale = 1.f / (nrm + EPS);
        sRed[0] = scale;
        const float qn = nrm * scale;   // ||q|| after normalization
        sNs[r0 + i] = qn * qn;          // q.q for later panels
      }
      __syncthreads();
      const float scale = sRed[0];
      for (int k = tid; k < DIM; k += NTHREADS)
        sU[i * DIM + k] *= scale;
      __syncthreads();
    }

    // ---- A5: write finished panel q rows to global ----
    {
      float4* g = (float4*)(Qg + (size_t)r0 * DIM);
      for (int t = tid; t < PANEL * DIM / 4; t += NTHREADS)
        g[t] = ((const float4*)sU)[t];
    }
    __syncthreads();
  }
}

extern "C" void kernel_launch(void* const* d_in, const int* in_sizes, int n_in,
                              void* d_out, int out_size, void* d_ws, size_t ws_size,
                              hipStream_t stream) {
  const float* V = (const float*)d_in[0];
  float*       Q = (float*)d_out;
  const int nbatch = in_sizes[0] / (NROW * DIM);   // 128 for the reference shapes
  hipLaunchKernelGGL(gram_schmidt_kernel, dim3(nbatch), dim3(NTHREADS), 0, stream,
                     V, Q);
}